// RoIAlign_71373766525293
// MI455X (gfx1250) — compile-verified
//
#include <hip/hip_runtime.h>

typedef float v2f __attribute__((ext_vector_type(2)));
typedef float v8f __attribute__((ext_vector_type(8)));

#define RA_N      2
#define RA_C      256
#define RA_H      200
#define RA_W      304
#define RA_OH     7
#define RA_OW     7
#define RA_SCALE  0.25f
#define NWAVES    4          // 4 waves/block, one channel per wave
#define TMPS      17         // padded LDS stride to avoid bank conflicts

// out(7x7) = Wy(7x28) * G(28x28) * Wx(28x7) per (roi, channel),
// computed with V_WMMA_F32_16X16X4_F32 tiles.
__global__ __launch_bounds__(NWAVES * 32)
void roialign_wmma_kernel(const float* __restrict__ feat,
                          const float* __restrict__ rois,
                          float* __restrict__ out)
{
    __shared__ int   s_yidx[28];
    __shared__ int   s_xidx[28];
    __shared__ float s_wy[28];                 // y-slot weights * 0.25 (SR^2 mean folded in)
    __shared__ float s_wx[28];                 // x-slot weights
    __shared__ float s_tmp[NWAVES][32][TMPS];  // stage-1 result, D-layout rows t=0..31
    __shared__ float s_out[NWAVES][56];        // 49-float output tile staging

    const int tid  = threadIdx.x;
    const int lane = tid & 31;
    const int wid  = tid >> 5;
    const int r    = blockIdx.x >> 6;          // 64 channel-groups per ROI
    const int cg   = blockIdx.x & 63;
    const int c    = cg * NWAVES + wid;

    // ---- ROI parameters (uniform across block -> scalar loads) ----
    const int   b    = (int)rois[r * 5 + 0];
    const float x1   = rois[r * 5 + 1] * RA_SCALE - 0.5f;
    const float y1   = rois[r * 5 + 2] * RA_SCALE - 0.5f;
    const float x2   = rois[r * 5 + 3] * RA_SCALE - 0.5f;
    const float y2   = rois[r * 5 + 4] * RA_SCALE - 0.5f;
    const float binw = (x2 - x1) * (1.0f / RA_OW);
    const float binh = (y2 - y1) * (1.0f / RA_OH);

    // ---- slot tables: t = 4*bin + 2*s + nb, sample coord = start + bin*(bin_idx + (s+0.5)/2)
    if (tid < 28) {
        const int   t  = tid;
        const int   nb = t & 1;
        const int   si = t >> 1;          // sample index 0..13
        const int   pb = si >> 1;         // bin 0..6
        const int   ss = si & 1;          // sub-sample 0..1
        const float g  = (float)pb + ((float)ss + 0.5f) * 0.5f;
        {   // y slots
            const float cy    = y1 + binh * g;
            const bool  valid = (cy > -1.0f) && (cy < (float)RA_H);
            float cc = cy < 0.0f ? 0.0f : cy;
            cc = cc > (float)(RA_H - 1) ? (float)(RA_H - 1) : cc;
            int i0 = (int)floorf(cc);
            int i1 = i0 + 1 < RA_H - 1 ? i0 + 1 : RA_H - 1;
            const float fr = cc - (float)i0;
            const float w  = nb ? fr : (1.0f - fr);
            s_yidx[t] = nb ? i1 : i0;
            s_wy[t]   = valid ? 0.25f * w : 0.0f;
        }
        {   // x slots
            const float cx    = x1 + binw * g;
            const bool  valid = (cx > -1.0f) && (cx < (float)RA_W);
            float cc = cx < 0.0f ? 0.0f : cx;
            cc = cc > (float)(RA_W - 1) ? (float)(RA_W - 1) : cc;
            int i0 = (int)floorf(cc);
            int i1 = i0 + 1 < RA_W - 1 ? i0 + 1 : RA_W - 1;
            const float fr = cc - (float)i0;
            const float w  = nb ? fr : (1.0f - fr);
            s_xidx[t] = nb ? i1 : i0;
            s_wx[t]   = valid ? w : 0.0f;
        }
    }
    __syncthreads();

    const int n  = lane & 15;   // WMMA column (stage1: row t base; stage2: pw)
    const int hi = lane >> 4;   // selects K pair {2hi, 2hi+1}
    const float* fbase = feat + ((size_t)b * RA_C + c) * (size_t)(RA_H * RA_W);

    // Per-lane weight pairs, loaded UNconditionally from a clamped index so the
    // per-chunk select is a register cndmask, not an exec-predicated ds_load.
    // (on = (kc == n) is only ever true for n < 7, so clamping n>=7 to 0 is safe.)
    const int   nn  = (n < 7) ? n : 0;
    const float bw0 = s_wx[4 * nn + 2 * hi];      // stage-1 B (Wx) pair
    const float bw1 = s_wx[4 * nn + 2 * hi + 1];
    const float aw0 = s_wy[4 * nn + 2 * hi];      // stage-2 A (Wy) pair
    const float aw1 = s_wy[4 * nn + 2 * hi + 1];

    // Preload all x-slot indices for this lane's K pairs into registers.
    int xoff[14];
#pragma unroll
    for (int kc = 0; kc < 7; ++kc) {
        xoff[2 * kc]     = s_xidx[4 * kc + 2 * hi];
        xoff[2 * kc + 1] = s_xidx[4 * kc + 2 * hi + 1];
    }

    // ---- Stage 1: tmp[t, pw] = sum_i G[t, i] * Wx[i, pw]  (Wx block-diagonal) ----
#pragma unroll
    for (int mt = 0; mt < 2; ++mt) {
        int t = 16 * mt + n;
        if (t > 27) t = 27;                       // clamp pad rows (results unused)
        const float* frow = fbase + (size_t)s_yidx[t] * RA_W;

        // Issue all 14 gathers first so they clause and overlap.
        float g[14];
#pragma unroll
        for (int j = 0; j < 14; ++j) g[j] = frow[xoff[j]];

        v8f d = {0.f, 0.f, 0.f, 0.f, 0.f, 0.f, 0.f, 0.f};
#pragma unroll
        for (int kc = 0; kc < 7; ++kc) {
            v2f A, B;
            A.x = g[2 * kc];
            A.y = g[2 * kc + 1];
            const bool on = (n == kc);            // Wx column kc only in chunk kc
            B.x = on ? bw0 : 0.0f;
            B.y = on ? bw1 : 0.0f;
            d = __builtin_amdgcn_wmma_f32_16x16x4_f32(false, A, false, B,
                                                      (short)0, d, false, false);
        }
        // D-layout: lane holds column n, rows m = 16*mt + 8*hi + e
#pragma unroll
        for (int e = 0; e < 8; ++e)
            s_tmp[wid][16 * mt + 8 * hi + e][n] = d[e];
    }
    __builtin_amdgcn_wave_barrier();  // LDS is wave-ordered; just pin scheduling

    // ---- Stage 2: out[ph, pw] = sum_t Wy[ph, t] * tmp[t, pw] ----
    v8f d2 = {0.f, 0.f, 0.f, 0.f, 0.f, 0.f, 0.f, 0.f};
#pragma unroll
    for (int kc = 0; kc < 7; ++kc) {
        const int t0 = 4 * kc + 2 * hi;
        v2f A, B;
        const bool on = (n == kc);                // Wy row kc only in chunk kc
        A.x = on ? aw0 : 0.0f;
        A.y = on ? aw1 : 0.0f;
        B.x = s_tmp[wid][t0][n];                  // tmp in B-layout (plain ds_loads)
        B.y = s_tmp[wid][t0 + 1][n];
        d2 = __builtin_amdgcn_wmma_f32_16x16x4_f32(false, A, false, B,
                                                   (short)0, d2, false, false);
    }

    // ---- stage out 7x7 tile, then coalesced global store ----
    if (hi == 0 && n < 7) {
#pragma unroll
        for (int e = 0; e < 7; ++e)
            s_out[wid][e * 7 + n] = d2[e];        // [ph][pw]
    }
    __builtin_amdgcn_wave_barrier();

    const size_t obase = ((size_t)r * RA_C + c) * (size_t)(RA_OH * RA_OW);
    for (int idx = lane; idx < RA_OH * RA_OW; idx += 32)
        out[obase + idx] = s_out[wid][idx];
}

extern "C" void kernel_launch(void* const* d_in, const int* in_sizes, int n_in,
                              void* d_out, int out_size, void* d_ws, size_t ws_size,
                              hipStream_t stream) {
    const float* feat = (const float*)d_in[0];
    const float* rois = (const float*)d_in[1];
    float*       out  = (float*)d_out;
    const int R = in_sizes[1] / 5;                 // 1024
    dim3 grid((unsigned)(R * (RA_C / NWAVES)));    // one block per (roi, 4-channel group)
    dim3 block(NWAVES * 32);
    hipLaunchKernelGGL(roialign_wmma_kernel, grid, block, 0, stream, feat, rois, out);
}